// LengthRegulator_65644280152320
// MI455X (gfx1250) — compile-verified
//
#include <hip/hip_runtime.h>
#include <stdint.h>

// LengthRegulator for MI455X (gfx1250).
// encoder_output: (32, 256, 512) f32 ; log_durations: (32, 256, 1) f32
// out: (32, 2048, 512) f32
//
// This is a zero-FLOP gather/expand: out[b,t,:] = enc[b, p(t), :] where p(t) is the
// phoneme whose cumulative-duration interval covers frame t (else zeros). Memory-bound:
// ~128 MiB stores + ~16 MiB HBM reads (encoder fits in L2) -> ~6 us floor at 23.3 TB/s.
// We use the CDNA5 async LDS copy path (GLOBAL_LOAD_ASYNC_TO_LDS_B128 /
// GLOBAL_STORE_ASYNC_FROM_LDS_B128, ASYNCcnt) for the row copies instead of WMMA:
// a one-hot einsum via V_WMMA_F32_16X16X4_F32 would add 17 GFLOP of useless f32
// matrix work and cannot beat the bandwidth roofline.

#define NB    32
#define NP    256
#define NC    512
#define NT    2048

// ---------------------------------------------------------------------------
// Kernel 1: durations -> cumsum -> (t -> phoneme) map, one block per batch.
// ---------------------------------------------------------------------------
__global__ __launch_bounds__(NP) void lr_scan_kernel(const float* __restrict__ log_dur,
                                                     int* __restrict__ map) {
  const int b = blockIdx.x;
  const int p = threadIdx.x;

  __shared__ int scan[NP];
  __shared__ int smap[NT];

  const float ld = log_dur[b * NP + p];
  int dur = 0;
  if (ld > 0.0f) dur = (int)floorf(exp2f(ld) + 1e-4f);  // matches floor(2^ld + 1e-4) * (ld>0)

  // Inclusive scan over 256 durations (Hillis-Steele, 8 steps).
  int val = dur;
  scan[p] = val;
  __syncthreads();
  for (int off = 1; off < NP; off <<= 1) {
    const int t = (p >= off) ? scan[p - off] : 0;
    __syncthreads();
    val += t;
    scan[p] = val;
    __syncthreads();
  }
  const int end   = val;        // exclusive end frame
  const int start = end - dur;  // start frame

  // Build t->p map in LDS: -1 means "no phoneme covers this frame".
  for (int i = p; i < NT; i += NP) smap[i] = -1;
  __syncthreads();
  const int end_c = end < NT ? end : NT;        // total can exceed NT in the worst case
  for (int t = start; t < end_c; ++t) smap[t] = p;  // intervals are disjoint -> race-free
  __syncthreads();
  for (int i = p; i < NT; i += NP) map[b * NT + i] = smap[i];
}

// ---------------------------------------------------------------------------
// Kernel 2: per-row gather-copy. 8 waves per block, one output row per wave.
// Real rows go through the gfx1250 async global<->LDS data mover; empty rows
// are zero-filled straight from VGPRs (no load side).
// ---------------------------------------------------------------------------
__global__ __launch_bounds__(256) void lr_copy_kernel(const float* __restrict__ enc,
                                                      const int* __restrict__ map,
                                                      float* __restrict__ out) {
  __shared__ float buf[8][NC];  // 16 KiB: one 2 KiB row staging slice per wave

  const int wave = threadIdx.x >> 5;    // 0..7
  const int lane = threadIdx.x & 31;
  const int row  = blockIdx.x * 8 + wave;  // flattened (b, t), 0..65535
  const int b    = row >> 11;              // row / 2048
  const int p    = map[row];               // wave-uniform

  float* __restrict__ dst = out + (size_t)row * NC;

  if (p < 0) {
    // Empty frame: write 2 KiB of zeros (4 x b128 per lane), fully coalesced.
    const float4 z = make_float4(0.0f, 0.0f, 0.0f, 0.0f);
    float4* d4 = (float4*)dst;
#pragma unroll
    for (int k = 0; k < 4; ++k) d4[lane + 32 * k] = z;
    return;
  }

  const float* __restrict__ src = enc + ((size_t)b * NP + p) * (size_t)NC;
  // addrspace(3) generic pointers carry the group-segment byte offset in the
  // low 32 bits; that is exactly the VDST/VSRC LDS address the async ops want.
  const uint32_t lds_base = (uint32_t)(uintptr_t)&buf[wave][0];

  // 4 x 512B async copies global -> LDS (32 lanes x 16B each), ASYNCcnt-tracked.
#pragma unroll
  for (int k = 0; k < 4; ++k) {
    const uint32_t ldsa = lds_base + (uint32_t)(lane + 32 * k) * 16u;
    const void*    ga   = (const char*)src + (size_t)(lane + 32 * k) * 16;
    asm volatile("global_load_async_to_lds_b128 %0, %1, off"
                 :: "v"(ldsa), "v"(ga)
                 : "memory");
  }
  asm volatile("s_wait_asynccnt 0" ::: "memory");

  // 4 x 512B async copies LDS -> global.
#pragma unroll
  for (int k = 0; k < 4; ++k) {
    const uint32_t ldsa = lds_base + (uint32_t)(lane + 32 * k) * 16u;
    void*          gd   = (char*)dst + (size_t)(lane + 32 * k) * 16;
    asm volatile("global_store_async_from_lds_b128 %0, %1, off"
                 :: "v"(gd), "v"(ldsa)
                 : "memory");
  }
  asm volatile("s_wait_asynccnt 0" ::: "memory");
}

// ---------------------------------------------------------------------------
extern "C" void kernel_launch(void* const* d_in, const int* in_sizes, int n_in,
                              void* d_out, int out_size, void* d_ws, size_t ws_size,
                              hipStream_t stream) {
  const float* enc  = (const float*)d_in[0];  // (32, 256, 512) f32
  const float* logd = (const float*)d_in[1];  // (32, 256, 1)  f32
  float*       out  = (float*)d_out;          // (32, 2048, 512) f32
  int*         map  = (int*)d_ws;             // 32*2048 ints = 256 KiB scratch

  lr_scan_kernel<<<NB, NP, 0, stream>>>(logd, map);
  lr_copy_kernel<<<(NB * NT) / 8, 256, 0, stream>>>(enc, map, out);
}